// GraphConvolution_67104569032788
// MI455X (gfx1250) — compile-verified
//
#include <hip/hip_runtime.h>

// GraphConvolution: out = relu(segment_sum(val * (x@W)[col], row))
// N=100000, E=1600000, D_IN=128, D_OUT=64 (derived from in_sizes at launch).

typedef __attribute__((ext_vector_type(2))) float v2f;
typedef __attribute__((ext_vector_type(8))) float v8f;

#define D_IN  128
#define D_OUT 64

// ---------------------------------------------------------------------------
// Kernel 1: transpose W [128,64] -> Wt [64,128] so B fragments are contiguous
// float2 loads along K.
// ---------------------------------------------------------------------------
__global__ __launch_bounds__(256) void gc_transpose_w(const float* __restrict__ w,
                                                      float* __restrict__ wt) {
    int i = blockIdx.x * blockDim.x + threadIdx.x;   // 0 .. 8191
    if (i < D_IN * D_OUT) {
        int k = i >> 6;        // 0..127
        int n = i & 63;        // 0..63
        wt[n * D_IN + k] = w[i];
    }
}

// ---------------------------------------------------------------------------
// Kernel 2: xw = x @ W via V_WMMA_F32_16X16X4_F32 (exact fp32 matrix path).
// One wave computes a 16x64 output tile: 4 accumulators (N tiles), K-loop of
// 32 steps of K=4. Fragment layout (f32 16x16x4, wave32):
//   A 16x4 : lane&15 = M, K = kbase + 2*(lane>>4) + reg   -> float2 per lane
//   B 4x16 : lane&15 = N, K = kbase + 2*(lane>>4) + reg   -> float2 per lane
//   C/D    : reg v: M = v + 8*(lane>>4), N = lane&15
// 100000 rows = 6250 tiles exactly -> EXEC all-1s inside active waves.
// ---------------------------------------------------------------------------
__global__ __launch_bounds__(256) void gc_gemm_wmma(const float* __restrict__ x,
                                                    const float* __restrict__ wt,
                                                    float* __restrict__ xw,
                                                    int ntiles) {
    int wave = (int)((blockIdx.x * blockDim.x + threadIdx.x) >> 5);
    int lane = threadIdx.x & 31;
    if (wave >= ntiles) return;                 // wave-uniform guard

    const int row0 = wave * 16;
    const int m    = lane & 15;                 // M for A / N-within-tile for B
    const int kh   = (lane >> 4) * 2;           // K sub-offset per lane half

    const float* __restrict__ arow = x  + (size_t)(row0 + m) * D_IN + kh;
    const float* __restrict__ b0p  = wt + (size_t)( 0 + m) * D_IN + kh;
    const float* __restrict__ b1p  = wt + (size_t)(16 + m) * D_IN + kh;
    const float* __restrict__ b2p  = wt + (size_t)(32 + m) * D_IN + kh;
    const float* __restrict__ b3p  = wt + (size_t)(48 + m) * D_IN + kh;

    v8f acc0 = {}, acc1 = {}, acc2 = {}, acc3 = {};

#pragma unroll 4
    for (int k = 0; k < D_IN; k += 4) {
        v2f a  = *(const v2f*)(arow + k);
        v2f b0 = *(const v2f*)(b0p + k);
        v2f b1 = *(const v2f*)(b1p + k);
        v2f b2 = *(const v2f*)(b2p + k);
        v2f b3 = *(const v2f*)(b3p + k);
        acc0 = __builtin_amdgcn_wmma_f32_16x16x4_f32(false, a, false, b0,
                                                     (short)0, acc0, false, false);
        acc1 = __builtin_amdgcn_wmma_f32_16x16x4_f32(false, a, false, b1,
                                                     (short)0, acc1, false, false);
        acc2 = __builtin_amdgcn_wmma_f32_16x16x4_f32(false, a, false, b2,
                                                     (short)0, acc2, false, false);
        acc3 = __builtin_amdgcn_wmma_f32_16x16x4_f32(false, a, false, b3,
                                                     (short)0, acc3, false, false);
    }

    // Store D: reg v -> row = row0 + v + 8*(lane>>4), col = nt*16 + (lane&15)
    const int rbase = row0 + (lane >> 4) * 8;
    const int n0    = lane & 15;
#pragma unroll
    for (int v = 0; v < 8; ++v) {
        float* o = xw + (size_t)(rbase + v) * D_OUT + n0;
        o[ 0] = acc0[v];
        o[16] = acc1[v];
        o[32] = acc2[v];
        o[48] = acc3[v];
    }
}

// ---------------------------------------------------------------------------
// Kernel 3: COO SpMM scatter. One wave per edge; edge scalars are
// wave-uniform (scalar loads), each lane covers 2 of the 64 output dims with
// a coalesced float2 gather (xw fits in the 192MB L2) + 2 fp32 atomics.
// ---------------------------------------------------------------------------
__global__ __launch_bounds__(256) void gc_spmm(const int* __restrict__ erow,
                                               const int* __restrict__ ecol,
                                               const float* __restrict__ eval,
                                               const float* __restrict__ xw,
                                               float* __restrict__ out,
                                               int E) {
    int e = blockIdx.x * 8 + (threadIdx.x >> 5);   // wave-uniform edge id
    if (e >= E) return;
    int lane = threadIdx.x & 31;

    int   r = erow[e];
    int   c = ecol[e];
    float v = eval[e];

    const v2f xv = *(const v2f*)(xw + (size_t)c * D_OUT + lane * 2);
    float* o = out + (size_t)r * D_OUT + lane * 2;
    atomicAdd(o + 0, v * xv.x);
    atomicAdd(o + 1, v * xv.y);
}

// ---------------------------------------------------------------------------
// Kernel 4: in-place ReLU, float4 vectorized.
// ---------------------------------------------------------------------------
__global__ __launch_bounds__(256) void gc_relu(float* __restrict__ out, int n4) {
    int i = blockIdx.x * blockDim.x + threadIdx.x;
    if (i < n4) {
        float4* p = (float4*)out;
        float4  v = p[i];
        v.x = fmaxf(v.x, 0.0f);
        v.y = fmaxf(v.y, 0.0f);
        v.z = fmaxf(v.z, 0.0f);
        v.w = fmaxf(v.w, 0.0f);
        p[i] = v;
    }
}

extern "C" void kernel_launch(void* const* d_in, const int* in_sizes, int n_in,
                              void* d_out, int out_size, void* d_ws, size_t ws_size,
                              hipStream_t stream) {
    const float* x    = (const float*)d_in[0];   // [N, 128]
    const int*   erow = (const int*)  d_in[1];   // [E]
    const int*   ecol = (const int*)  d_in[2];   // [E]
    const float* ev   = (const float*)d_in[3];   // [E]
    const float* w    = (const float*)d_in[4];   // [128, 64]
    float*       out  = (float*)d_out;           // [N, 64]

    const int N = in_sizes[0] / D_IN;
    const int E = in_sizes[1];

    // Scratch layout: xw [N*64 floats] then Wt [64*128 floats]
    float* xw = (float*)d_ws;
    float* wt = xw + (size_t)N * D_OUT;

    // out = 0 (memset is graph-capture safe)
    hipMemsetAsync(d_out, 0, (size_t)out_size * sizeof(float), stream);

    // W -> Wt
    gc_transpose_w<<<(D_IN * D_OUT + 255) / 256, 256, 0, stream>>>(w, wt);

    // xw = x @ W  (one wave per 16-row tile; 8 waves per block)
    const int ntiles = N / 16;                       // 6250, exact
    gc_gemm_wmma<<<(ntiles + 7) / 8, 256, 0, stream>>>(x, wt, xw, ntiles);

    // scatter-accumulate over edges (one wave per edge)
    long long spmm_threads = (long long)E * 32;
    gc_spmm<<<(int)((spmm_threads + 255) / 256), 256, 0, stream>>>(
        erow, ecol, ev, xw, out, E);

    // relu
    const int n4 = (N * D_OUT) / 4;
    gc_relu<<<(n4 + 255) / 256, 256, 0, stream>>>(out, n4);
}